// RegionContrastiveLoss_14705968022088
// MI455X (gfx1250) — compile-verified
//
#include <hip/hip_runtime.h>
#include <hip/hip_bf16.h>
#include <math.h>

typedef __attribute__((ext_vector_type(2))) float v2f;
typedef __attribute__((ext_vector_type(8))) float v8f;

#define N_BATCH   2048
#define TWO_N     4096
#define THREE_N   6144
#define FEAT      128
#define INV_T     2.0f          // 1 / TEMPERATURE
#define COL_SPLIT 8
#define COLS_PER_BLOCK 512      // TWO_N / COL_SPLIT
#define ROWS_PER_BLOCK 32
#define LDS_STRIDE 132          // 128 + 4 pad: (132 % 64 == 4) -> conflict-free f2 reads

// ---------------------------------------------------------------------------
// Kernel 1: row-normalize concat(anchor, pos, neg) -> zn [3N x 128]
// ---------------------------------------------------------------------------
__global__ __launch_bounds__(256) void nt_normalize(const float* __restrict__ anchor,
                                                    const float* __restrict__ pos,
                                                    const float* __restrict__ neg,
                                                    float* __restrict__ zn) {
  const int wid = threadIdx.x >> 5, lane = threadIdx.x & 31;
  const int row = blockIdx.x * 8 + wid;            // grid = 3N/8 = 768
  const float* src;
  if (row < N_BATCH)      src = anchor + (size_t)row * FEAT;
  else if (row < TWO_N)   src = pos    + (size_t)(row - N_BATCH) * FEAT;
  else                    src = neg    + (size_t)(row - TWO_N)   * FEAT;
  float4 x = ((const float4*)src)[lane];           // 4 elems/lane * 32 lanes = 128
  float ss = x.x * x.x + x.y * x.y + x.z * x.z + x.w * x.w;
  #pragma unroll
  for (int off = 1; off < 32; off <<= 1) ss += __shfl_xor(ss, off, 32);
  const float inv = 1.0f / fmaxf(sqrtf(ss), 1e-8f);
  float4 o; o.x = x.x * inv; o.y = x.y * inv; o.z = x.z * inv; o.w = x.w * inv;
  ((float4*)(zn + (size_t)row * FEAT))[lane] = o;
}

// ---------------------------------------------------------------------------
// Kernel 2: positives p[i] = <zn_i, zn_{2N+i}> * (1/T)
// ---------------------------------------------------------------------------
__global__ __launch_bounds__(256) void nt_positives(const float* __restrict__ zn,
                                                    float* __restrict__ p) {
  const int wid = threadIdx.x >> 5, lane = threadIdx.x & 31;
  const int i = blockIdx.x * 8 + wid;              // grid = N/8 = 256
  float4 a = ((const float4*)(zn + (size_t)i * FEAT))[lane];
  float4 b = ((const float4*)(zn + (size_t)(TWO_N + i) * FEAT))[lane];
  float d = a.x * b.x + a.y * b.y + a.z * b.z + a.w * b.w;
  #pragma unroll
  for (int off = 1; off < 32; off <<= 1) d += __shfl_xor(d, off, 32);
  if (lane == 0) p[i] = d * INV_T;
}

// ---------------------------------------------------------------------------
// Kernel 3: WMMA f32 sim block + masked exp accumulation.
// Block = 256 threads / 8 waves. Block covers 32 rows x 512 cols.
// Wave w: tr = w&1 selects 16-row half; tc = w>>1 selects 16-col quarter
//         of the current 64-col tile.
// Fragment layout (ISA 32-bit 16x4): VGPR v, lane L -> element [L%16][v+2*(L>=16)]
// => both A and B fragments are a float2 load at [tileRow + L%16][kbase + 2*(L>=16)].
// Two independent accumulators (even/odd k-steps) break the WMMA RAW chain.
// ---------------------------------------------------------------------------
__global__ __launch_bounds__(256) void nt_simexp(const float* __restrict__ zn,
                                                 float* __restrict__ pLo,
                                                 float* __restrict__ pHi) {
  __shared__ float ldsA[ROWS_PER_BLOCK * LDS_STRIDE];
  __shared__ float ldsB[64 * LDS_STRIDE];
  __shared__ float ldsAcc[4][2][16][2];            // [tc][tr][row16][lo/hi]

  const int t = threadIdx.x, lane = t & 31, wid = t >> 5;
  const int m16 = lane & 15, hic = lane >> 4;
  const int tr = wid & 1, tc = wid >> 1;
  const int R0 = blockIdx.x * ROWS_PER_BLOCK;      // grid.x = 4096/32 = 128
  const int C0 = blockIdx.y * COLS_PER_BLOCK;      // grid.y = 8

  // Load A tile: 32 rows x 128 floats = 1024 float4, 4 per thread.
  #pragma unroll
  for (int it = 0; it < 4; ++it) {
    int f = t + it * 256;
    int row = f >> 5, c4 = f & 31;
    float4 v = ((const float4*)(zn + (size_t)(R0 + row) * FEAT))[c4];
    *(float4*)&ldsA[row * LDS_STRIDE + c4 * 4] = v;
  }
  // Prefetch first B tile (one 128B cacheline per thread covers the 32KB tile).
  __builtin_prefetch(zn + (size_t)(C0 + (t >> 2)) * FEAT + (t & 3) * 32, 0, 1);

  float accLo[8], accHi[8];
  #pragma unroll
  for (int v = 0; v < 8; ++v) { accLo[v] = 0.0f; accHi[v] = 0.0f; }

  for (int ct = 0; ct < 8; ++ct) {
    __syncthreads();
    // Load B tile: rows = columns C0+ct*64 .. +63 of zn; 2048 float4, 8/thread.
    #pragma unroll
    for (int it = 0; it < 8; ++it) {
      int f = t + it * 256;
      int row = f >> 5, c4 = f & 31;
      float4 v = ((const float4*)(zn + (size_t)(C0 + ct * 64 + row) * FEAT))[c4];
      *(float4*)&ldsB[row * LDS_STRIDE + c4 * 4] = v;
    }
    __syncthreads();
    // Prefetch next B tile while this one computes.
    if (ct < 7)
      __builtin_prefetch(zn + (size_t)(C0 + (ct + 1) * 64 + (t >> 2)) * FEAT + (t & 3) * 32, 0, 1);

    v8f c0 = {0.f, 0.f, 0.f, 0.f, 0.f, 0.f, 0.f, 0.f};
    v8f c1 = {0.f, 0.f, 0.f, 0.f, 0.f, 0.f, 0.f, 0.f};
    const float* Ab = &ldsA[(tr * 16 + m16) * LDS_STRIDE + 2 * hic];
    const float* Bb = &ldsB[(tc * 16 + m16) * LDS_STRIDE + 2 * hic];
    #pragma unroll
    for (int kk = 0; kk < 32; kk += 2) {           // K = 128 in steps of 4, 2 at a time
      v2f a0 = *(const v2f*)(Ab + kk * 4);
      v2f b0 = *(const v2f*)(Bb + kk * 4);
      v2f a1 = *(const v2f*)(Ab + kk * 4 + 4);
      v2f b1 = *(const v2f*)(Bb + kk * 4 + 4);
      c0 = __builtin_amdgcn_wmma_f32_16x16x4_f32(false, a0, false, b0,
                                                 (short)0, c0, false, false);
      c1 = __builtin_amdgcn_wmma_f32_16x16x4_f32(false, a1, false, b1,
                                                 (short)0, c1, false, false);
    }

    // D layout: c[v] at lane L -> row (v + 8*hic), col (L%16).
    const int rbase = R0 + tr * 16 + 8 * hic;
    const int cc = C0 + ct * 64 + tc * 16 + m16;
    #pragma unroll
    for (int v = 0; v < 8; ++v) {
      const int r = rbase + v;
      const float e = expf((c0[v] + c1[v]) * INV_T);
      if (r < N_BATCH) {
        if (cc != r && cc != r + N_BATCH) accHi[v] += e;
      } else {
        const int j = r - N_BATCH;
        if (cc < j) accLo[v] += e;                       // spills to logits row r-1
        else if (cc > j && cc != j + N_BATCH) accHi[v] += e;
      }
    }
  }

  // Deterministic reduction across the 16 lanes of each half-wave.
  #pragma unroll
  for (int v = 0; v < 8; ++v) {
    #pragma unroll
    for (int off = 1; off < 16; off <<= 1) {
      accLo[v] += __shfl_xor(accLo[v], off, 32);
      accHi[v] += __shfl_xor(accHi[v], off, 32);
    }
  }
  if (m16 == 0) {
    #pragma unroll
    for (int v = 0; v < 8; ++v) {
      const int rr = v + 8 * hic;
      ldsAcc[tc][tr][rr][0] = accLo[v];
      ldsAcc[tc][tr][rr][1] = accHi[v];
    }
  }
  __syncthreads();
  if (t < ROWS_PER_BLOCK) {                        // fixed-order combine of 4 col-quarters
    const int trr = t >> 4, rr = t & 15;
    float lo = 0.0f, hi = 0.0f;
    #pragma unroll
    for (int q = 0; q < 4; ++q) {
      lo += ldsAcc[q][trr][rr][0];
      hi += ldsAcc[q][trr][rr][1];
    }
    const int r = R0 + trr * 16 + rr;
    pLo[(size_t)blockIdx.y * TWO_N + r] = lo;
    pHi[(size_t)blockIdx.y * TWO_N + r] = hi;
  }
}

// ---------------------------------------------------------------------------
// Kernel 4: fixed-order reduction of col-split partials.
// ---------------------------------------------------------------------------
__global__ __launch_bounds__(256) void nt_reduce_partials(const float* __restrict__ pLo,
                                                          const float* __restrict__ pHi,
                                                          float* __restrict__ LoT,
                                                          float* __restrict__ HiT) {
  const int r = blockIdx.x * 256 + threadIdx.x;    // grid = 16
  if (r < TWO_N) {
    float lo = 0.0f, hi = 0.0f;
    #pragma unroll
    for (int s = 0; s < COL_SPLIT; ++s) {
      lo += pLo[(size_t)s * TWO_N + r];
      hi += pHi[(size_t)s * TWO_N + r];
    }
    LoT[r] = lo;
    HiT[r] = hi;
  }
}

// ---------------------------------------------------------------------------
// Kernel 5: final loss. Exact regrouped logsumexp:
//  i < N     : log(HiT[i] + 2*e^{p_i})
//  k < N-1   : log(e^{p_k} + HiT[N+k] + LoT[N+k+1])
//  k = N-1   : log(e^{p_{N-1}} + HiT[2N-1] + sum_i e^{p_i})
//  loss = (sum_logs - 2*sum_i p_i) / 2N
// ---------------------------------------------------------------------------
__global__ __launch_bounds__(256) void nt_final(const float* __restrict__ p,
                                                const float* __restrict__ LoT,
                                                const float* __restrict__ HiT,
                                                float* __restrict__ out) {
  __shared__ float red[256];
  __shared__ float sExpP, sP;
  const int t = threadIdx.x;

  float eSum = 0.0f, pSum = 0.0f;
  for (int i = t; i < N_BATCH; i += 256) { const float pv = p[i]; eSum += expf(pv); pSum += pv; }
  red[t] = eSum; __syncthreads();
  for (int s = 128; s > 0; s >>= 1) { if (t < s) red[t] += red[t + s]; __syncthreads(); }
  if (t == 0) sExpP = red[0];
  __syncthreads();
  red[t] = pSum; __syncthreads();
  for (int s = 128; s > 0; s >>= 1) { if (t < s) red[t] += red[t + s]; __syncthreads(); }
  if (t == 0) sP = red[0];
  __syncthreads();

  float term = 0.0f;
  for (int i = t; i < N_BATCH; i += 256)
    term += logf(HiT[i] + LoT[i] + 2.0f * expf(p[i]));   // LoT[i] == 0 for i < N
  for (int k = t; k < N_BATCH; k += 256) {
    const float ep = expf(p[k]);
    const float tail = (k < N_BATCH - 1) ? LoT[N_BATCH + k + 1] : sExpP;
    term += logf(ep + HiT[N_BATCH + k] + tail);
  }
  red[t] = term; __syncthreads();
  for (int s = 128; s > 0; s >>= 1) { if (t < s) red[t] += red[t + s]; __syncthreads(); }
  if (t == 0) out[0] = (red[0] - 2.0f * sP) / (float)TWO_N;
}

// ---------------------------------------------------------------------------
extern "C" void kernel_launch(void* const* d_in, const int* in_sizes, int n_in,
                              void* d_out, int out_size, void* d_ws, size_t ws_size,
                              hipStream_t stream) {
  const float* anchor = (const float*)d_in[0];
  const float* pos    = (const float*)d_in[1];
  const float* neg    = (const float*)d_in[2];
  float* out = (float*)d_out;

  // Workspace carve-up (floats): zn | p | pLo | pHi | LoT | HiT  (~3.45 MB)
  float* ws  = (float*)d_ws;
  float* zn  = ws;                                 // 3N*128 = 786432
  float* p   = zn  + (size_t)THREE_N * FEAT;       // 2048
  float* pLo = p   + N_BATCH;                      // 8*4096
  float* pHi = pLo + (size_t)COL_SPLIT * TWO_N;    // 8*4096
  float* LoT = pHi + (size_t)COL_SPLIT * TWO_N;    // 4096
  float* HiT = LoT + TWO_N;                        // 4096

  nt_normalize<<<dim3(THREE_N / 8), dim3(256), 0, stream>>>(anchor, pos, neg, zn);
  nt_positives<<<dim3(N_BATCH / 8), dim3(256), 0, stream>>>(zn, p);
  nt_simexp<<<dim3(TWO_N / ROWS_PER_BLOCK, COL_SPLIT), dim3(256), 0, stream>>>(zn, pLo, pHi);
  nt_reduce_partials<<<dim3(TWO_N / 256), dim3(256), 0, stream>>>(pLo, pHi, LoT, HiT);
  nt_final<<<dim3(1), dim3(256), 0, stream>>>(p, LoT, HiT, out);
}